// MultiHeadAttention_4750233829652
// MI455X (gfx1250) — compile-verified
//
#include <hip/hip_runtime.h>
#include <hip/hip_bf16.h>
#include <math.h>

typedef float v2f __attribute__((ext_vector_type(2)));
typedef float v8f __attribute__((ext_vector_type(8)));
typedef int   v4i_ __attribute__((__vector_size__(4 * sizeof(int))));

#define B_    2
#define S_    4096
#define D_    512
#define H_    8
#define DK_   64
#define SCALE_ 0.125f          // 1/sqrt(64)
#define PAD_  68               // 64 + 4: ds_load_b64 B-operands are conflict-free

#if __has_builtin(__builtin_amdgcn_global_load_async_to_lds_b128)
#define HAVE_ASYNC_LDS 1
#else
#define HAVE_ASYNC_LDS 0
#endif

__device__ __forceinline__ void async_cp16(const float* g, float* l) {
#if HAVE_ASYNC_LDS
    __builtin_amdgcn_global_load_async_to_lds_b128(
        (__attribute__((address_space(1))) v4i_*)(float*)g,
        (__attribute__((address_space(3))) v4i_*)l,
        /*offset=*/0, /*cpol=*/0);
#endif
}

__device__ __forceinline__ void async_wait0() {
#if __has_builtin(__builtin_amdgcn_s_wait_asynccnt)
    __builtin_amdgcn_s_wait_asynccnt(0);
#elif HAVE_ASYNC_LDS
    asm volatile("s_wait_asynccnt 0x0" ::: "memory");
#endif
}

// Exact fp32 WMMA: D(16x16,f32) = A(16x4,f32) * B(4x16,f32) + C
// A/B: lane = row/col, float2 at k + 2*(lane>>4).  C/D: M = r + 8*(l>>4), N = l&15.
__device__ __forceinline__ v8f wmma4(v2f a, v2f b, v8f c) {
    return __builtin_amdgcn_wmma_f32_16x16x4_f32(
        false, a, false, b, (short)0, c, false, false);
}

// ---------------------------------------------------------------------------
// K1: fused QKV projection, y = x @ W.T, stored head-major [B,H,S,DK].
// Block: 8 waves share one head's 64 weight rows (staged per 64-col chunk).
// grid = 3 * 8 heads * 64 mt-groups = 1536 blocks
// ---------------------------------------------------------------------------
__global__ __launch_bounds__(256) void k_qkv(
    const float* __restrict__ q_in, const float* __restrict__ k_in,
    const float* __restrict__ v_in,
    const float* __restrict__ Wq, const float* __restrict__ Wk,
    const float* __restrict__ Wv,
    float* __restrict__ Qh, float* __restrict__ Kh, float* __restrict__ Vh)
{
    __shared__ float bufW[64][PAD_];
#if HAVE_ASYNC_LDS
    __shared__ float bufA[64 * 64];
#endif
    const int t = threadIdx.x, waveId = t >> 5, lane = t & 31;
    const int bi = blockIdx.x;
    const int which = bi / 512;
    const int h     = (bi % 512) / 64;
    const int mt    = (bi % 64) * 8 + waveId;

    const float* X = (which == 0) ? q_in : (which == 1) ? k_in : v_in;
    const float* W = (which == 0) ? Wq   : (which == 1) ? Wk   : Wv;
    float*       O = (which == 0) ? Qh   : (which == 1) ? Kh   : Vh;

    const int m0 = mt * 16, l15 = lane & 15, koff = (lane >> 4) * 2;
    const float* xrow   = X + (size_t)(m0 + l15) * D_ + koff;
    const float* wpanel = W + (size_t)h * DK_ * D_;     // 64 rows of W

    v8f acc0 = {}, acc1 = {}, acc2 = {}, acc3 = {};
    for (int kb = 0; kb < D_; kb += 64) {
        __syncthreads();
#if HAVE_ASYNC_LDS
#pragma unroll
        for (int i = 0; i < 4; ++i) {
            const int li4 = t + i * 256;                 // 1024 float4 = 16KB
            const int row = li4 >> 4, cg = li4 & 15;
            async_cp16(wpanel + (size_t)row * D_ + kb + cg * 4, bufA + li4 * 4);
        }
        async_wait0();
        __syncthreads();
#pragma unroll
        for (int i = 0; i < 16; ++i) {
            const int li = t + i * 256;
            bufW[li >> 6][li & 63] = bufA[li];
        }
#else
#pragma unroll
        for (int i = 0; i < 4; ++i) {
            const int li4 = t + i * 256;
            const int row = li4 >> 4, cg = (li4 & 15) * 4;
            const float4 d = *(const float4*)(wpanel + (size_t)row * D_ + kb + cg);
            bufW[row][cg + 0] = d.x; bufW[row][cg + 1] = d.y;
            bufW[row][cg + 2] = d.z; bufW[row][cg + 3] = d.w;
        }
#endif
        __syncthreads();
#pragma unroll
        for (int kk = 0; kk < 64; kk += 4) {
            v2f a = *(const v2f*)(xrow + kb + kk);
            acc0 = wmma4(a, *(const v2f*)&bufW[ 0 + l15][kk + koff], acc0);
            acc1 = wmma4(a, *(const v2f*)&bufW[16 + l15][kk + koff], acc1);
            acc2 = wmma4(a, *(const v2f*)&bufW[32 + l15][kk + koff], acc2);
            acc3 = wmma4(a, *(const v2f*)&bufW[48 + l15][kk + koff], acc3);
        }
    }
#pragma unroll
    for (int r = 0; r < 8; ++r) {
        const int row = m0 + r + 8 * (lane >> 4);
        const int b = row / S_, s = row % S_;
        float* orow = O + (((size_t)(b * H_ + h) * S_) + s) * DK_;
        orow[ 0 + l15] = acc0[r];
        orow[16 + l15] = acc1[r];
        orow[32 + l15] = acc2[r];
        orow[48 + l15] = acc3[r];
    }
}

// ---------------------------------------------------------------------------
// K2: raw scores = scale * Q @ K^T into the attn region of d_out.
// Block: 8 waves (8 q-tiles) share one 64-row K panel (contiguous 16KB).
// grid = 16 bh * 32 qt-groups * 64 col-groups = 32768 blocks
// ---------------------------------------------------------------------------
__global__ __launch_bounds__(256) void k_scores(
    const float* __restrict__ Qh, const float* __restrict__ Kh,
    float* __restrict__ attn)
{
    __shared__ float bufK[64][PAD_];
#if HAVE_ASYNC_LDS
    __shared__ float bufA[64 * 64];
#endif
    const int t = threadIdx.x, waveId = t >> 5, lane = t & 31;
    const int bi  = blockIdx.x;
    const int bh  = bi / 2048;
    const int qtg = (bi / 64) % 32;
    const int ng  = bi % 64;
    const int qt  = qtg * 8 + waveId;

    const int q0 = qt * 16, c0 = ng * 64;
    const int l15 = lane & 15, koff = (lane >> 4) * 2;
    const float* qrow   = Qh + ((size_t)bh * S_ + q0 + l15) * DK_ + koff;
    const float* kpanel = Kh + ((size_t)bh * S_ + c0) * DK_;   // contiguous 4096 f32

#if HAVE_ASYNC_LDS
#pragma unroll
    for (int i = 0; i < 4; ++i) {
        const int li4 = t + i * 256;
        async_cp16(kpanel + (size_t)li4 * 4, bufA + li4 * 4);
    }
    async_wait0();
    __syncthreads();
#pragma unroll
    for (int i = 0; i < 16; ++i) {
        const int li = t + i * 256;
        bufK[li >> 6][li & 63] = bufA[li];
    }
#else
#pragma unroll
    for (int i = 0; i < 4; ++i) {
        const int li4 = t + i * 256;
        const int row = li4 >> 4, cg = (li4 & 15) * 4;
        const float4 d = *(const float4*)(kpanel + (size_t)li4 * 4);
        bufK[row][cg + 0] = d.x; bufK[row][cg + 1] = d.y;
        bufK[row][cg + 2] = d.z; bufK[row][cg + 3] = d.w;
    }
#endif
    __syncthreads();

    v8f acc0 = {}, acc1 = {}, acc2 = {}, acc3 = {};
#pragma unroll
    for (int kk = 0; kk < DK_; kk += 4) {
        v2f a = *(const v2f*)(qrow + kk);
        acc0 = wmma4(a, *(const v2f*)&bufK[ 0 + l15][kk + koff], acc0);
        acc1 = wmma4(a, *(const v2f*)&bufK[16 + l15][kk + koff], acc1);
        acc2 = wmma4(a, *(const v2f*)&bufK[32 + l15][kk + koff], acc2);
        acc3 = wmma4(a, *(const v2f*)&bufK[48 + l15][kk + koff], acc3);
    }
#pragma unroll
    for (int r = 0; r < 8; ++r) {
        const int row = q0 + r + 8 * (lane >> 4);
        float* prow = attn + ((size_t)bh * S_ + row) * S_;
        prow[c0 +  0 + l15] = acc0[r] * SCALE_;
        prow[c0 + 16 + l15] = acc1[r] * SCALE_;
        prow[c0 + 32 + l15] = acc2[r] * SCALE_;
        prow[c0 + 48 + l15] = acc3[r] * SCALE_;
    }
}

// ---------------------------------------------------------------------------
// K3: in-place masked softmax over each 4096-wide attn row (register-resident).
// ---------------------------------------------------------------------------
__global__ __launch_bounds__(256) void k_softmax(
    float* __restrict__ attn, const int* __restrict__ mask)
{
    const int row = blockIdx.x;               // in [0, B*H*S)
    const int b   = row / (H_ * S_);
    float* p = attn + (size_t)row * S_;
    const int* mrow = mask + b * S_;
    const int t = threadIdx.x;

    float v[16];
    float mx = -INFINITY;
#pragma unroll
    for (int i = 0; i < 16; ++i) {
        const int c = t + i * 256;
        float x = p[c];
        x = (mrow[c] == 0) ? -1e9f : x;
        v[i] = x;
        mx = fmaxf(mx, x);
    }
    __shared__ float red[256];
    red[t] = mx;
    __syncthreads();
    for (int s = 128; s > 0; s >>= 1) {
        if (t < s) red[t] = fmaxf(red[t], red[t + s]);
        __syncthreads();
    }
    mx = red[0];
    __syncthreads();

    float sum = 0.f;
#pragma unroll
    for (int i = 0; i < 16; ++i) {
        v[i] = expf(v[i] - mx);
        sum += v[i];
    }
    red[t] = sum;
    __syncthreads();
    for (int s = 128; s > 0; s >>= 1) {
        if (t < s) red[t] += red[t + s];
        __syncthreads();
    }
    const float inv = 1.0f / red[0];
#pragma unroll
    for (int i = 0; i < 16; ++i) p[t + i * 256] = v[i] * inv;
}

// ---------------------------------------------------------------------------
// K4: ctx = attn @ V.  Block: 8 waves (128 q-rows, one head) share each
// 64-key V chunk, staged transposed bufT[dim][key] so B-operands are v2f.
// grid = 16 bh * 32 qt-groups = 512 blocks
// ---------------------------------------------------------------------------
__global__ __launch_bounds__(256) void k_av(
    const float* __restrict__ attn, const float* __restrict__ Vh,
    float* __restrict__ ctx)
{
    __shared__ float bufT[64][PAD_];          // bufT[dim][key] = V[key][dim]
#if HAVE_ASYNC_LDS
    __shared__ float bufA[64 * 64];
#endif
    const int t = threadIdx.x, waveId = t >> 5, lane = t & 31;
    const int bh = blockIdx.x >> 5;
    const int qt = (blockIdx.x & 31) * 8 + waveId;
    const int b = bh / H_, h = bh % H_;

    const int q0 = qt * 16, l15 = lane & 15, koff = (lane >> 4) * 2;
    const float* prow = attn + ((size_t)bh * S_ + q0 + l15) * S_ + koff;
    const float* vb   = Vh + (size_t)bh * S_ * DK_;

    v8f acc0 = {}, acc1 = {}, acc2 = {}, acc3 = {};
    for (int c = 0; c < S_; c += 64) {
        __syncthreads();
#if HAVE_ASYNC_LDS
#pragma unroll
        for (int i = 0; i < 4; ++i) {
            const int li4 = t + i * 256;
            async_cp16(vb + (size_t)c * DK_ + li4 * 4, bufA + li4 * 4);
        }
        async_wait0();
        __syncthreads();
#pragma unroll
        for (int i = 0; i < 16; ++i) {
            const int li = t + i * 256;
            bufT[li & 63][li >> 6] = bufA[li];   // transpose: [dim][key]
        }
#else
#pragma unroll
        for (int i = 0; i < 4; ++i) {
            const int li4 = t + i * 256;
            const int key = li4 >> 4, d4 = (li4 & 15) * 4;
            const float4 d = *(const float4*)(vb + (size_t)c * DK_ + li4 * 4);
            bufT[d4 + 0][key] = d.x; bufT[d4 + 1][key] = d.y;
            bufT[d4 + 2][key] = d.z; bufT[d4 + 3][key] = d.w;
        }
#endif
        __syncthreads();
#pragma unroll
        for (int kk = 0; kk < 64; kk += 4) {
            v2f a = *(const v2f*)(prow + c + kk);
            const int k0 = kk + koff;
            acc0 = wmma4(a, *(const v2f*)&bufT[ 0 + l15][k0], acc0);
            acc1 = wmma4(a, *(const v2f*)&bufT[16 + l15][k0], acc1);
            acc2 = wmma4(a, *(const v2f*)&bufT[32 + l15][k0], acc2);
            acc3 = wmma4(a, *(const v2f*)&bufT[48 + l15][k0], acc3);
        }
    }
#pragma unroll
    for (int r = 0; r < 8; ++r) {
        const int s = q0 + r + 8 * (lane >> 4);
        float* crow = ctx + ((size_t)(b * S_ + s)) * D_ + h * DK_;
        crow[ 0 + l15] = acc0[r];
        crow[16 + l15] = acc1[r];
        crow[32 + l15] = acc2[r];
        crow[48 + l15] = acc3[r];
    }
}

// ---------------------------------------------------------------------------
// K5: out = ctx @ W_o.T + b_o.  Block: 8 waves share one 64-row W_o panel.
// grid = 64 mt-groups * 8 col-groups = 512 blocks
// ---------------------------------------------------------------------------
__global__ __launch_bounds__(256) void k_oproj(
    const float* __restrict__ ctx, const float* __restrict__ Wo,
    const float* __restrict__ bo, float* __restrict__ out)
{
    __shared__ float bufW[64][PAD_];
#if HAVE_ASYNC_LDS
    __shared__ float bufA[64 * 64];
#endif
    const int t = threadIdx.x, waveId = t >> 5, lane = t & 31;
    const int ng = blockIdx.x & 7;
    const int mt = (blockIdx.x >> 3) * 8 + waveId;

    const int m0 = mt * 16, n0 = ng * 64;
    const int l15 = lane & 15, koff = (lane >> 4) * 2;
    const float* xrow   = ctx + (size_t)(m0 + l15) * D_ + koff;
    const float* wpanel = Wo + (size_t)n0 * D_;

    v8f acc0 = {}, acc1 = {}, acc2 = {}, acc3 = {};
    for (int kb = 0; kb < D_; kb += 64) {
        __syncthreads();
#if HAVE_ASYNC_LDS
#pragma unroll
        for (int i = 0; i < 4; ++i) {
            const int li4 = t + i * 256;
            const int row = li4 >> 4, cg = li4 & 15;
            async_cp16(wpanel + (size_t)row * D_ + kb + cg * 4, bufA + li4 * 4);
        }
        async_wait0();
        __syncthreads();
#pragma unroll
        for (int i = 0; i < 16; ++i) {
            const int li = t + i * 256;
            bufW[li >> 6][li & 63] = bufA[li];
        }
#else
#pragma unroll
        for (int i = 0; i < 4; ++i) {
            const int li4 = t + i * 256;
            const int row = li4 >> 4, cg = (li4 & 15) * 4;
            const float4 d = *(const float4*)(wpanel + (size_t)row * D_ + kb + cg);
            bufW[row][cg + 0] = d.x; bufW[row][cg + 1] = d.y;
            bufW[row][cg + 2] = d.z; bufW[row][cg + 3] = d.w;
        }
#endif
        __syncthreads();
#pragma unroll
        for (int kk = 0; kk < 64; kk += 4) {
            v2f a = *(const v2f*)(xrow + kb + kk);
            acc0 = wmma4(a, *(const v2f*)&bufW[ 0 + l15][kk + koff], acc0);
            acc1 = wmma4(a, *(const v2f*)&bufW[16 + l15][kk + koff], acc1);
            acc2 = wmma4(a, *(const v2f*)&bufW[32 + l15][kk + koff], acc2);
            acc3 = wmma4(a, *(const v2f*)&bufW[48 + l15][kk + koff], acc3);
        }
    }
#pragma unroll
    for (int r = 0; r < 8; ++r) {
        const int row = m0 + r + 8 * (lane >> 4);
        float* orow = out + (size_t)row * D_;
        orow[n0 +  0 + l15] = acc0[r] + bo[n0 +  0 + l15];
        orow[n0 + 16 + l15] = acc1[r] + bo[n0 + 16 + l15];
        orow[n0 + 32 + l15] = acc2[r] + bo[n0 + 32 + l15];
        orow[n0 + 48 + l15] = acc3[r] + bo[n0 + 48 + l15];
    }
}

// ---------------------------------------------------------------------------
extern "C" void kernel_launch(void* const* d_in, const int* in_sizes, int n_in,
                              void* d_out, int out_size, void* d_ws, size_t ws_size,
                              hipStream_t stream)
{
    const float* q    = (const float*)d_in[0];
    const float* k    = (const float*)d_in[1];
    const float* v    = (const float*)d_in[2];
    const int*   mask = (const int*)  d_in[3];
    const float* Wq   = (const float*)d_in[4];
    const float* Wk   = (const float*)d_in[5];
    const float* Wv   = (const float*)d_in[6];
    const float* Wo   = (const float*)d_in[7];
    const float* bo   = (const float*)d_in[8];

    float* out  = (float*)d_out;                      // [B, S, D]
    float* attn = out + (size_t)B_ * S_ * D_;         // [B, H, S, S]

    const size_t sz = (size_t)B_ * S_ * D_;
    float* Qh  = (float*)d_ws;                        // [B, H, S, DK]
    float* Kh  = Qh + sz;
    float* Vh  = Kh + sz;
    float* ctx = Vh + sz;                             // [B, S, D]

    k_qkv<<<dim3(3 * H_ * 64), dim3(256), 0, stream>>>(
        q, k, v, Wq, Wk, Wv, Qh, Kh, Vh);

    k_scores<<<dim3(B_ * H_ * 32 * 64), dim3(256), 0, stream>>>(Qh, Kh, attn);

    k_softmax<<<dim3(B_ * H_ * S_), dim3(256), 0, stream>>>(attn, mask);

    k_av<<<dim3(B_ * H_ * 32), dim3(256), 0, stream>>>(attn, Vh, ctx);

    k_oproj<<<dim3(64 * 8), dim3(256), 0, stream>>>(ctx, Wo, bo, out);
}